// Disc_edge4_15573551415688
// MI455X (gfx1250) — compile-verified
//
#include <hip/hip_runtime.h>
#include <stdint.h>

typedef float v2f __attribute__((ext_vector_type(2)));
typedef float v8f __attribute__((ext_vector_type(8)));

// B is stored pair-interleaved in LDS: Bp[p*LDP + 2*n + q] = B[2p+q][n]
// (p = k-pair index, n = output column, q = 0/1). Each lane's WMMA B operand
// is then one aligned 8-byte LDS load. LDP=160 -> lane halves (p vs p+1) hit
// disjoint bank halves (160 mod 64 == 32).
#define LDP 160

// ---------------------------------------------------------------------------
// Per-wave GEMM building block: 16 rows x K  @  K x 64  ->  four 16x16 f32
// accumulator tiles, using V_WMMA_F32_16X16X4_F32 (full fp32 precision).
// A layout (16x4, MxK): lane l: m=l&15, kh=l>>4; a = {A[m][2kh], A[m][2kh+1]}
// B layout (4x16, KxN): lane l: n=l&15, kh=l>>4; b = {B[2kh][n], B[2kh+1][n]}
// C/D layout (16x16):   VGPR r, lane l: row = r + 8*(l>>4), col = l&15
// ---------------------------------------------------------------------------
__device__ __forceinline__ void wmma_gemm_16xKx64(
    const float* __restrict__ A, int ldA, int rowBase,
    const float* __restrict__ Bp, int K,
    v8f acc[4], int lane) {
  const int m  = lane & 15;
  const int kh = (lane >> 4) & 1;
  for (int k = 0; k < K; k += 4) {
    v2f a;
    a.x = A[(rowBase + m) * ldA + k + 2 * kh];
    a.y = A[(rowBase + m) * ldA + k + 2 * kh + 1];
    const float* brow = Bp + ((k >> 1) + kh) * LDP + 2 * m;
#pragma unroll
    for (int t = 0; t < 4; ++t) {
      v2f bb = *reinterpret_cast<const v2f*>(&brow[t * 32]);
      acc[t] = __builtin_amdgcn_wmma_f32_16x16x4_f32(
          false, a, false, bb, (short)0, acc[t], false, false);
    }
  }
}

__device__ __forceinline__ void copy_tile_g2l(float* dst, int ldDst,
                                              const float* __restrict__ src,
                                              int rows, int cols, int tid, int nthr) {
  const int total4 = (rows * cols) >> 2;   // cols is a multiple of 4
  for (int idx = tid; idx < total4; idx += nthr) {
    int off = idx * 4;
    int r = off / cols, c = off % cols;
    float4 v = reinterpret_cast<const float4*>(src)[idx];
    *reinterpret_cast<float4*>(&dst[r * ldDst + c]) = v;
  }
}

// src: rows x 64 row-major contiguous -> pair-interleaved LDS layout
__device__ __forceinline__ void copy_B_interleaved(float* dstP,
                                                   const float* __restrict__ src,
                                                   int rows, int tid, int nthr) {
  const int total4 = rows * 16;
  for (int idx = tid; idx < total4; idx += nthr) {
    int off = idx * 4;
    int r = off >> 6, c = off & 63;
    float4 v = reinterpret_cast<const float4*>(src)[idx];
    float* d = dstP + (r >> 1) * LDP + (r & 1) + 2 * c;
    d[0] = v.x; d[2] = v.y; d[4] = v.z; d[6] = v.w;
  }
}

// ---------------------------------------------------------------------------
// Generic GEMM: out[b] = act( concat(A0[b],A1[b]) @ B + bias )
// Used for Y1/Y2 precompute (no bias/relu) and node update (bias+relu).
// K0, K1 are multiples of 32, so 32-wide K-chunks never straddle the concat.
// ---------------------------------------------------------------------------
__global__ __launch_bounds__(256) void gnn_gemm_kernel(
    const float* __restrict__ A0, int K0,
    const float* __restrict__ A1, int K1,
    const float* __restrict__ Bm,
    const float* __restrict__ bias, int relu,
    float* __restrict__ out) {
  __shared__ float As[128 * 36];   // 128 x 32 chunk, padded stride
  __shared__ float Bp[16 * LDP];   // 32 x 64 chunk, pair-interleaved
  __shared__ float bs[64];

  const int tid = threadIdx.x;
  const int b   = blockIdx.x;
  const int K   = K0 + K1;
  const int w = tid >> 5, lane = tid & 31;

  if (tid < 64) bs[tid] = bias ? bias[tid] : 0.0f;

  v8f acc[4];
  v8f z = {0.f, 0.f, 0.f, 0.f, 0.f, 0.f, 0.f, 0.f};
  acc[0] = z; acc[1] = z; acc[2] = z; acc[3] = z;

  for (int kc = 0; kc < K; kc += 32) {
    const float* src; int scol, sld;
    if (kc < K0) { src = A0 + (size_t)b * 128 * K0; scol = kc;      sld = K0; }
    else         { src = A1 + (size_t)b * 128 * K1; scol = kc - K0; sld = K1; }
    __syncthreads();   // protect LDS reuse across chunks
    for (int idx = tid; idx < (128 * 32) / 4; idx += 256) {
      int off = idx * 4, r = off >> 5, c = off & 31;
      float4 v = *reinterpret_cast<const float4*>(&src[(size_t)r * sld + scol + c]);
      *reinterpret_cast<float4*>(&As[r * 36 + c]) = v;
    }
    copy_B_interleaved(Bp, Bm + kc * 64, 32, tid, 256);
    __syncthreads();
    wmma_gemm_16xKx64(As, 36, 16 * w, Bp, 32, acc, lane);
  }

  const int m = lane & 15, kh = lane >> 4;
#pragma unroll
  for (int t = 0; t < 4; ++t) {
    int n = t * 16 + m;
#pragma unroll
    for (int r = 0; r < 8; ++r) {
      int j = 16 * w + 8 * kh + r;
      float v = acc[t][r] + bs[n];
      if (relu) v = fmaxf(v, 0.0f);
      out[((size_t)b * 128 + j) * 64 + n] = v;
    }
  }
}

// ---------------------------------------------------------------------------
// Fused edge update. One block per (b,i): 128 edge rows x 64 out channels.
//   e_new[j,:] = relu( e[j]@We3 + Y1[i] + Y2[j] + be ) * adj[i,j]
// The 32KB e-tile is staged into LDS with CDNA5 async global->LDS B128 copies
// (ASYNCcnt), bypassing VGPRs. Also produces colsum_j (deterministic LDS
// tree), deg, agg = colsum/deg, and raw colsum for the edge-mean readout.
// In-place safe: the whole e tile is staged in LDS before any store.
// ---------------------------------------------------------------------------
__global__ __launch_bounds__(256) void gnn_edge_kernel(
    const int*   __restrict__ adj,
    const float* __restrict__ e_in,
    const float* __restrict__ W3p,   // We[2fn:2fn+feIn, :]
    const float* __restrict__ be,
    const float* __restrict__ Y1,
    const float* __restrict__ Y2,
    float* __restrict__ e_out,
    float* __restrict__ agg,
    float* __restrict__ aggraw,
    int feIn) {
  __shared__ float Ae[128 * 68];
  __shared__ float Bp[32 * LDP];
  __shared__ float y1s[64], bes[64], adjf[128];
  __shared__ float psum[64 * 16];
  __shared__ float degs;

  const int tid = threadIdx.x;
  const int b = blockIdx.x >> 7, i = blockIdx.x & 127;
  const int ldA = feIn + 4;

  // Async e-tile copy: global -> LDS, B128 per lane, tracked by ASYNCcnt.
  {
    const float* src = e_in + (size_t)(b * 128 + i) * 128 * feIn;
    const int total4 = (128 * feIn) >> 2;
    for (int idx = tid; idx < total4; idx += 256) {
      int off = idx * 4;
      int r = off / feIn, c = off % feIn;
      // Flat LDS-aperture addresses carry the LDS byte offset in addr[31:0].
      uint32_t ldsoff = (uint32_t)(uintptr_t)&Ae[r * ldA + c];
      const float* gp = src + off;
      asm volatile("global_load_async_to_lds_b128 %0, %1, off"
                   :: "v"(ldsoff), "v"(gp) : "memory");
    }
  }
  copy_B_interleaved(Bp, W3p, feIn, tid, 256);
  if (tid < 128) adjf[tid] = (float)adj[(size_t)(b * 128 + i) * 128 + tid];
  if (tid < 64) { y1s[tid] = Y1[(b * 128 + i) * 64 + tid]; bes[tid] = be[tid]; }
  asm volatile("s_wait_asynccnt 0" ::: "memory");
  __syncthreads();

  if (tid == 0) {
    float s = 0.0f;
    for (int j = 0; j < 128; ++j) s += adjf[j];
    degs = fmaxf(s, 1.0f);
  }

  const int w = tid >> 5, lane = tid & 31;
  v8f acc[4];
  v8f z = {0.f, 0.f, 0.f, 0.f, 0.f, 0.f, 0.f, 0.f};
  acc[0] = z; acc[1] = z; acc[2] = z; acc[3] = z;
  wmma_gemm_16xKx64(Ae, ldA, 16 * w, Bp, feIn, acc, lane);

  const float* y2p = Y2 + (size_t)b * 128 * 64;   // hot in 192MB L2 (512KB total)
  const int m = lane & 15, kh = lane >> 4;
#pragma unroll
  for (int t = 0; t < 4; ++t) {
    int n = t * 16 + m;
    float base = y1s[n] + bes[n];
    float cs = 0.0f;
#pragma unroll
    for (int r = 0; r < 8; ++r) {
      int j = 16 * w + 8 * kh + r;
      float v = acc[t][r] + base + y2p[j * 64 + n];
      v = fmaxf(v, 0.0f) * adjf[j];
      e_out[(((size_t)(b * 128 + i)) * 128 + j) * 64 + n] = v;
      cs += v;
    }
    psum[n * 16 + w * 2 + kh] = cs;   // unique writer per (n,w,kh): deterministic
  }
  __syncthreads();
  if (tid < 64) {
    float s = 0.0f;
#pragma unroll
    for (int p = 0; p < 16; ++p) s += psum[tid * 16 + p];
    aggraw[(b * 128 + i) * 64 + tid] = s;         // for edge-mean readout
    agg[(b * 128 + i) * 64 + tid]    = s / degs;  // for node update
  }
}

// ---------------------------------------------------------------------------
// Readout: g[b] = sum_i colsum[b,i] / (N*N), then 64->64->64->1 MLP. Tiny.
// ---------------------------------------------------------------------------
__global__ __launch_bounds__(256) void gnn_readout_kernel(
    const float* __restrict__ aggraw,
    const float* __restrict__ W1, const float* __restrict__ b1,
    const float* __restrict__ W2, const float* __restrict__ b2,
    const float* __restrict__ W3, const float* __restrict__ b3,
    float* __restrict__ out) {
  __shared__ float g[16 * 64];
  __shared__ float h[16 * 64];
  const int tid = threadIdx.x;

  for (int t = tid; t < 1024; t += 256) {
    int b = t >> 6, f = t & 63;
    float s = 0.0f;
    for (int i = 0; i < 128; ++i) s += aggraw[(b * 128 + i) * 64 + f];
    g[t] = s * (1.0f / 16384.0f);
  }
  __syncthreads();
  for (int t = tid; t < 1024; t += 256) {
    int b = t >> 6, f = t & 63;
    float s = b1[f];
    for (int k = 0; k < 64; ++k) s += g[b * 64 + k] * W1[k * 64 + f];
    h[t] = fmaxf(s, 0.0f);
  }
  __syncthreads();
  for (int t = tid; t < 1024; t += 256) {
    int b = t >> 6, f = t & 63;
    float s = b2[f];
    for (int k = 0; k < 64; ++k) s += h[b * 64 + k] * W2[k * 64 + f];
    g[t] = fmaxf(s, 0.0f);
  }
  __syncthreads();
  if (tid < 16) {
    float s = b3[0];
    for (int k = 0; k < 64; ++k) s += g[tid * 64 + k] * W3[k];
    out[tid] = s;
  }
}

// ---------------------------------------------------------------------------
extern "C" void kernel_launch(void* const* d_in, const int* in_sizes, int n_in,
                              void* d_out, int out_size, void* d_ws, size_t ws_size,
                              hipStream_t stream) {
  const int*   adj = (const int*)d_in[0];
  const float* x0  = (const float*)d_in[1];
  const float* e0  = (const float*)d_in[2];
  const float* We[3] = {(const float*)d_in[3], (const float*)d_in[7],  (const float*)d_in[11]};
  const float* be[3] = {(const float*)d_in[4], (const float*)d_in[8],  (const float*)d_in[12]};
  const float* Wn[3] = {(const float*)d_in[5], (const float*)d_in[9],  (const float*)d_in[13]};
  const float* bn[3] = {(const float*)d_in[6], (const float*)d_in[10], (const float*)d_in[14]};
  const float* W1 = (const float*)d_in[15]; const float* b1 = (const float*)d_in[16];
  const float* W2 = (const float*)d_in[17]; const float* b2 = (const float*)d_in[18];
  const float* W3 = (const float*)d_in[19]; const float* b3 = (const float*)d_in[20];

  // Workspace layout (floats): ~70.3 MB total
  float* ws     = (float*)d_ws;
  float* e_buf  = ws;                                    // 16*128*128*64 = 16,777,216
  float* xbufA  = e_buf + (size_t)16 * 128 * 128 * 64;   // 131072
  float* xbufB  = xbufA + 16 * 128 * 64;
  float* Y1     = xbufB + 16 * 128 * 64;
  float* Y2     = Y1 + 16 * 128 * 64;
  float* aggb   = Y2 + 16 * 128 * 64;
  float* aggraw = aggb + 16 * 128 * 64;

  const int fns[3] = {32, 64, 64};
  const int fes[3] = {32, 64, 64};
  const float* xl = x0;
  const float* el = e0;

  for (int l = 0; l < 3; ++l) {
    const int fn = fns[l], fe = fes[l];
    // Y1 = X @ We[0:fn], Y2 = X @ We[fn:2fn]
    gnn_gemm_kernel<<<dim3(16), dim3(256), 0, stream>>>(
        xl, fn, nullptr, 0, We[l], nullptr, 0, Y1);
    gnn_gemm_kernel<<<dim3(16), dim3(256), 0, stream>>>(
        xl, fn, nullptr, 0, We[l] + fn * 64, nullptr, 0, Y2);
    // fused edge update (+colsum/deg/agg); layers 1,2 run in place on e_buf
    gnn_edge_kernel<<<dim3(2048), dim3(256), 0, stream>>>(
        adj, el, We[l] + 2 * fn * 64, be[l], Y1, Y2, e_buf, aggb, aggraw, fe);
    el = e_buf;
    if (l < 2) {  // layer 2's node update is dead code in the reference
      float* xo = (l == 0) ? xbufA : xbufB;
      gnn_gemm_kernel<<<dim3(16), dim3(256), 0, stream>>>(
          xl, fn, aggb, 64, Wn[l], bn[l], 1, xo);
      xl = xo;
    }
  }
  gnn_readout_kernel<<<dim3(1), dim3(256), 0, stream>>>(
      aggraw, W1, b1, W2, b2, W3, b3, (float*)d_out);

  (void)in_sizes; (void)n_in; (void)out_size; (void)ws_size;
}